// NDCGLoss_51754355917566
// MI455X (gfx1250) — compile-verified
//
#include <hip/hip_runtime.h>
#include <hip/hip_bf16.h>

// ---------------------------------------------------------------------------
// NDCG for N~16.7M items on MI455X (gfx1250, wave32).
//   1) keys = desc-sortable u32 of output; vals = gain bits
//   2) stable LSD radix sort, 8-bit digits, 4 passes (pair sort)
//   3) dcg = sum(val_sorted[i] / log2(i+2))   -- WMMA f32 16x16x4 reduction
//   4) radix sort gains descending (key-only, 4 passes), zk = same dot
//   5) out = dcg / zk
// Scatter tiles are staged into LDS by the Tensor Data Mover with HW padding
// (pad 1 dword per 64) which doubles as the LDS bank de-swizzle.
// ---------------------------------------------------------------------------

#define THREADS  256
#define TILE     4096
#define RADIX    256
#define NPASS    4
#define REDB     512                    // reduction blocks
#define PTILE    (TILE + (TILE >> 6))   // padded tile (1 dword pad per 64)
#define STHREADS 32                     // scatter block = one wave
#define SIPT     (TILE / STHREADS)      // 128 items per scatter thread
#define PADM     33                     // padded thread-stride of rank matrix

typedef float v2f  __attribute__((ext_vector_type(2)));
typedef float v8f  __attribute__((ext_vector_type(8)));
typedef unsigned int u32x4 __attribute__((ext_vector_type(4)));
typedef int   i32x4 __attribute__((ext_vector_type(4)));
typedef int   i32x8 __attribute__((ext_vector_type(8)));

#if __has_builtin(__builtin_amdgcn_wmma_f32_16x16x4_f32)
#define HAVE_WMMA 1
#endif
#if __has_builtin(__builtin_amdgcn_tensor_load_to_lds) && \
    __has_builtin(__builtin_amdgcn_s_wait_tensorcnt)
#define HAVE_TDM 1
#endif
#if __has_builtin(__builtin_amdgcn_rcpf)
#define FASTRCP(x) __builtin_amdgcn_rcpf(x)
#else
#define FASTRCP(x) (1.0f / (x))
#endif

// Compile-time probes (stderr only; non-fatal).
#if defined(__AMDGCN__)
#ifdef HAVE_TDM
#if __clang_major__ >= 23
#warning "CDNA5 probe: TDM available, 6-arg tensor_load_to_lds (clang>=23)"
#else
#warning "CDNA5 probe: TDM available, 5-arg tensor_load_to_lds (clang<23)"
#endif
#else
#warning "CDNA5 probe: tensor_load_to_lds builtin MISSING (plain-load staging)"
#endif
#endif

// Descending-sortable key for float bits; involution (its own inverse).
__device__ __forceinline__ unsigned int fkey(unsigned int u) {
  return (u & 0x80000000u) ? u : (~u & 0x7fffffffu);
}
// LDS de-swizzle: +1 dword pad every 64 dwords (== TDM pad_interval=5).
__device__ __forceinline__ int padi(int i) { return i + (i >> 6); }

#ifdef HAVE_TDM
// One TDM load of TILE dwords (one 4096-elem row) into LDS with padding.
// D# encoding per CDNA5 ISA.
__device__ __forceinline__ void tdm_load_tile(const unsigned int* gsrc,
                                              unsigned int lds_off) {
  unsigned long long ga = (unsigned long long)(size_t)gsrc;
  u32x4 g0;
  g0[0] = 1u;                                     // count=1, user descriptor
  g0[1] = lds_off;                                // LDS byte address
  g0[2] = (unsigned int)ga;                       // global addr [31:0]
  g0[3] = (unsigned int)(ga >> 32) | 0x80000000u; // addr[56:32] | type=2
  i32x8 g1 = {};
  // data_size=4B(2), pad_enable=1, pad_interval=5 (64 dwords), pad_amount=0 (1 dword)
  g1[0] = (int)((2u << 16) | (1u << 20) | (5u << 22));
  g1[1] = (int)((unsigned)(TILE & 0xFFFF) << 16);      // tensor_dim0 lo16
  g1[2] = (int)(((unsigned)TILE >> 16) | (1u << 16));  // tensor_dim0 hi, tensor_dim1=1
  g1[3] = (int)((unsigned)TILE << 16);                 // tile_dim0 = TILE
  g1[4] = 1;                                           // tile_dim1 = 1
  g1[5] = TILE;                                        // tensor_dim0_stride lo
  i32x4 z4 = {};
#if __clang_major__ >= 23
  i32x8 z8 = {};
  __builtin_amdgcn_tensor_load_to_lds(g0, g1, z4, z4, z8, 0);
#else
  __builtin_amdgcn_tensor_load_to_lds(g0, g1, z4, z4, 0);
#endif
}
#endif

// --------------------------- key/gain generation ---------------------------
__global__ __launch_bounds__(THREADS)
void k_make(const float* __restrict__ outp, const float* __restrict__ lab,
            unsigned int* __restrict__ K0, unsigned int* __restrict__ V0, int n) {
  const float LMAX = 13.982028007507324f, LMIN = -11.172813415527344f;
  for (int i = blockIdx.x * blockDim.x + threadIdx.x; i < n;
       i += gridDim.x * blockDim.x) {
    float l = lab[i] * (LMAX - LMIN) + LMIN;
    float g = exp2f(l) - 1.0f;
    V0[i] = __float_as_uint(g);
    K0[i] = fkey(__float_as_uint(outp[i]));
  }
}

__global__ __launch_bounds__(THREADS)
void k_gkeys(const unsigned int* __restrict__ V, unsigned int* __restrict__ K, int n) {
  for (int i = blockIdx.x * blockDim.x + threadIdx.x; i < n;
       i += gridDim.x * blockDim.x)
    K[i] = fkey(V[i]);
}

// ------------------------------- histogram ---------------------------------
__global__ __launch_bounds__(THREADS)
void k_hist(const unsigned int* __restrict__ keys, unsigned int* __restrict__ counts,
            int n, int shift, int numTiles) {
  __shared__ unsigned int sh[8 * RADIX];   // per-wave counters (8KB)
  const int tid = threadIdx.x, tile = blockIdx.x;
  const int base = tile * TILE;
  for (int i = tid; i < 8 * RADIX; i += THREADS) sh[i] = 0u;
  __syncthreads();
  if (base + TILE + tid < n)               // warm next tile (global_prefetch_b8)
    __builtin_prefetch(&keys[base + TILE + tid], 0, 0);
  const int w = tid >> 5;
  for (int i = tid; i < TILE; i += THREADS) {
    int idx = base + i;
    if (idx < n) {
      unsigned int d = (keys[idx] >> shift) & 255u;
      atomicAdd(&sh[w * RADIX + d], 1u);
    }
  }
  __syncthreads();
  if (tid < RADIX) {
    unsigned int s = 0;
    for (int ww = 0; ww < 8; ww++) s += sh[ww * RADIX + tid];
    counts[(size_t)tid * numTiles + tile] = s;   // digit-major layout
  }
}

// ------------------ hierarchical exclusive scan (3 kernels) ----------------
__global__ __launch_bounds__(THREADS)
void k_scan_blk(unsigned int* __restrict__ data, unsigned int* __restrict__ bsums,
                int L) {
  __shared__ unsigned int loc[THREADS];
  const int tid = threadIdx.x, b = blockIdx.x;
  const int base = b * 1024 + tid * 4;
  unsigned int s = 0;
#pragma unroll
  for (int k = 0; k < 4; k++) s += (base + k < L) ? data[base + k] : 0u;
  loc[tid] = s;
  __syncthreads();
  for (int off = 1; off < THREADS; off <<= 1) {
    unsigned int v = (tid >= off) ? loc[tid - off] : 0u;
    __syncthreads();
    loc[tid] += v;
    __syncthreads();
  }
  if (tid == 0) bsums[b] = loc[THREADS - 1];
  unsigned int run = (tid == 0) ? 0u : loc[tid - 1];
#pragma unroll
  for (int k = 0; k < 4; k++) {
    int idx = base + k;
    if (idx < L) { unsigned int v = data[idx]; data[idx] = run; run += v; }
  }
}

__global__ __launch_bounds__(THREADS)
void k_scan_top(unsigned int* __restrict__ data, int L) {   // generic, 1 block
  __shared__ unsigned int sums[THREADS];
  const int tid = threadIdx.x;
  const int chunk = (L + THREADS - 1) / THREADS;
  int s0 = tid * chunk, s1 = s0 + chunk;
  if (s1 > L) s1 = L;
  unsigned int s = 0;
  for (int i = s0; i < s1; i++) s += data[i];
  sums[tid] = s;
  __syncthreads();
  for (int off = 1; off < THREADS; off <<= 1) {
    unsigned int v = (tid >= off) ? sums[tid - off] : 0u;
    __syncthreads();
    sums[tid] += v;
    __syncthreads();
  }
  unsigned int run = (tid == 0) ? 0u : sums[tid - 1];
  for (int i = s0; i < s1; i++) {
    unsigned int v = data[i];
    data[i] = run;
    run += v;
  }
}

__global__ __launch_bounds__(THREADS)
void k_scan_add(unsigned int* __restrict__ data, const unsigned int* __restrict__ bsums,
                int L) {
  const int tid = threadIdx.x, b = blockIdx.x;
  const unsigned int add = bsums[b];
  const int base = b * 1024 + tid * 4;
#pragma unroll
  for (int k = 0; k < 4; k++) {
    int idx = base + k;
    if (idx < L) data[idx] += add;
  }
}

// ------------------------- stable radix scatter ----------------------------
// One wave per block; thread t owns the contiguous chunk [t*128, t*128+128) of
// the tile, so (thread, item) order == element order => stable.
template <bool HASVAL>
__global__ __launch_bounds__(STHREADS)
void k_scatter(const unsigned int* __restrict__ kin,
               const unsigned int* __restrict__ vin,
               unsigned int* __restrict__ kout,
               unsigned int* __restrict__ vout,
               const unsigned int* __restrict__ offsets,
               int n, int shift, int numTiles) {
  __shared__ unsigned int shk[PTILE];            // 16.6 KB (padded tile)
  __shared__ unsigned int mat[RADIX * PADM];     // 33 KB  [digit][thread], pad 33
  __shared__ unsigned int ssc[STHREADS];
  __shared__ unsigned int gbase[RADIX];
  __shared__ unsigned int dbase[RADIX];

  const int tid = threadIdx.x, tile = blockIdx.x;
  const int base = tile * TILE;

  // ---- stage key tile into LDS (TDM with HW padding, else plain loads)
  bool staged = false;
#ifdef HAVE_TDM
  if (base + TILE <= n) {
    if (tid == 0) tdm_load_tile(kin + base, (unsigned int)(size_t)(&shk[0]));
    __builtin_amdgcn_s_wait_tensorcnt(0);
    staged = true;
  }
#endif
  if (!staged) {
    for (int i = tid; i < TILE; i += STHREADS) {
      int idx = base + i;
      shk[padi(i)] = (idx < n) ? kin[idx] : 0u;
    }
  }
  __syncthreads();

  // ---- per-thread digit counts over the contiguous chunk
  for (int d = 0; d < RADIX; d++) mat[d * PADM + tid] = 0u;
  const int tbeg = tid * SIPT;
  for (int k = 0; k < SIPT; k++) {
    int i = tbeg + k;
    if (base + i < n) {
      unsigned int d = (shk[padi(i)] >> shift) & 255u;
      mat[d * PADM + tid]++;
    }
  }
  __syncthreads();

  // ---- flat exclusive scan over logical [digit][thread] order (stable ranks)
  unsigned int s = 0;
  for (int q = 0; q < 8; q++)
    for (int t2 = 0; t2 < STHREADS; t2++)
      s += mat[(tid * 8 + q) * PADM + t2];
  ssc[tid] = s;
  __syncthreads();
  for (int off = 1; off < STHREADS; off <<= 1) {
    unsigned int v = (tid >= off) ? ssc[tid - off] : 0u;
    __syncthreads();
    ssc[tid] += v;
    __syncthreads();
  }
  unsigned int run = (tid == 0) ? 0u : ssc[tid - 1];
  for (int q = 0; q < 8; q++)
    for (int t2 = 0; t2 < STHREADS; t2++) {
      unsigned int* p = &mat[(tid * 8 + q) * PADM + t2];
      unsigned int v = *p;
      *p = run;
      run += v;
    }
  __syncthreads();

  for (int d = tid; d < RADIX; d += STHREADS) {
    dbase[d] = mat[d * PADM];                            // prefix of digits < d
    gbase[d] = offsets[(size_t)d * numTiles + tile];     // global base
  }
  __syncthreads();
  for (int d = 0; d < RADIX; d++)
    mat[d * PADM + tid] += gbase[d] - dbase[d];          // absolute cursors
  __syncthreads();

  // ---- stable scatter (values read directly; thread-sequential => cache hits)
  for (int k = 0; k < SIPT; k++) {
    int i = tbeg + k;
    int idx = base + i;
    if (idx < n) {
      unsigned int kk = shk[padi(i)];
      unsigned int d = (kk >> shift) & 255u;
      unsigned int pos = mat[d * PADM + tid]++;
      kout[pos] = kk;
      if constexpr (HASVAL) vout[pos] = vin[idx];
    }
  }
}

// -------------------- disc-weighted dot via WMMA reduction ------------------
// Wave handles 64 consecutive elements/iter via scalar b32 loads (clamped
// index, no divergence, no vector-load->WMMA folding: that pattern crashed
// LLVM 23 ISel). B = all-ones 4x16, C accumulates row sums in every column of
// D; grand total = sum of D[:,0] = v8f comps of lanes 0 and 16. Any
// element->A-slot assignment is valid for a pure sum.
template <bool FROMKEY>
__global__ __launch_bounds__(THREADS)
void k_dot(const unsigned int* __restrict__ data, double* __restrict__ partial,
           int n, int iters) {
  const int tid = threadIdx.x, lane = tid & 31, wid = tid >> 5;
  const int gw = blockIdx.x * 8 + wid;
  const int totalW = gridDim.x * 8;
#ifdef HAVE_WMMA
  v8f acc = {0.f, 0.f, 0.f, 0.f, 0.f, 0.f, 0.f, 0.f};
  const v2f ones = {1.0f, 1.0f};
#else
  float facc = 0.0f;
#endif
  for (int it = 0; it < iters; ++it) {       // uniform trip count (EXEC all-1)
    const int basei = (it * totalW + gw) * 64 + lane * 2;  // fits in i32
    float a0, a1;
    {
      int i = basei;
      bool ok = i < n;
      unsigned int u = data[ok ? i : 0];     // clamped load, select result
      if (FROMKEY) u = fkey(u);
      float w = FASTRCP(__log2f((float)i + 2.0f));
      a0 = ok ? __uint_as_float(u) * w : 0.0f;
    }
    {
      int i = basei + 1;
      bool ok = i < n;
      unsigned int u = data[ok ? i : 0];
      if (FROMKEY) u = fkey(u);
      float w = FASTRCP(__log2f((float)i + 2.0f));
      a1 = ok ? __uint_as_float(u) * w : 0.0f;
    }
#ifdef HAVE_WMMA
    // Opaque register barrier: forces a0/a1 into plain VGPRs so ISel cannot
    // fold load results straight into the WMMA source operands.
    __asm__ volatile("" : "+v"(a0), "+v"(a1));
    v2f a;
    a[0] = a0;
    a[1] = a1;
    acc = __builtin_amdgcn_wmma_f32_16x16x4_f32(false, a, false, ones,
                                                (short)0, acc, false, false);
#else
    facc += a0 + a1;
#endif
  }
  double wsum;
#ifdef HAVE_WMMA
  float s = 0.f;
#pragma unroll
  for (int k = 0; k < 8; k++) s += acc[k];
  float s0 = __shfl(s, 0, 32);    // rows 0..7  of column 0
  float s16 = __shfl(s, 16, 32);  // rows 8..15 of column 0
  wsum = (double)s0 + (double)s16;
#else
  float s = facc;
  for (int off = 16; off > 0; off >>= 1) s += __shfl_xor(s, off, 32);
  wsum = (double)s;
#endif
  __shared__ double sm[8];
  if (lane == 0) sm[wid] = wsum;
  __syncthreads();
  if (tid == 0) {
    double t = 0.0;
    for (int q = 0; q < 8; q++) t += sm[q];
    partial[blockIdx.x] = t;
  }
}

__global__ void k_final(const double* __restrict__ pa, const double* __restrict__ pb,
                        int m, float* __restrict__ out) {
  if (threadIdx.x == 0 && blockIdx.x == 0) {
    double a = 0.0, b = 0.0;
    for (int i = 0; i < m; i++) { a += pa[i]; b += pb[i]; }   // fixed order
    out[0] = (float)(a / b);
  }
}

// --------------------------------- host ------------------------------------
static inline void run_scan(unsigned int* counts, unsigned int* bsums, int L,
                            hipStream_t stream) {
  int nb = (L + 1023) / 1024;
  k_scan_blk<<<nb, THREADS, 0, stream>>>(counts, bsums, L);
  k_scan_top<<<1, THREADS, 0, stream>>>(bsums, nb);
  k_scan_add<<<nb, THREADS, 0, stream>>>(counts, bsums, L);
}

extern "C" void kernel_launch(void* const* d_in, const int* in_sizes, int n_in,
                              void* d_out, int out_size, void* d_ws, size_t ws_size,
                              hipStream_t stream) {
  const int n = in_sizes[0];
  const float* outp = (const float*)d_in[0];
  const float* lab  = (const float*)d_in[1];
  float* out = (float*)d_out;

  const int numTiles = (n + TILE - 1) / TILE;
  const int L = RADIX * numTiles;
  const size_t nAl = ((size_t)n + 1023) & ~(size_t)1023;
  unsigned int* K0 = (unsigned int*)d_ws;
  unsigned int* K1 = K0 + nAl;
  unsigned int* V0 = K1 + nAl;
  unsigned int* V1 = V0 + nAl;
  unsigned int* counts = V1 + nAl;
  const size_t cntLen = (size_t)L;
  unsigned int* bsums = counts + cntLen;               // cntLen multiple of 256
  const size_t nbAl = (((size_t)(L + 1023) / 1024) + 1) & ~(size_t)1;
  double* pA = (double*)(bsums + nbAl);
  double* pB = pA + REDB;

  int mb = (n + THREADS * 8 - 1) / (THREADS * 8);
  if (mb < 1) mb = 1;
  if (mb > 16384) mb = 16384;

  // 1) build sort keys + gains
  k_make<<<mb, THREADS, 0, stream>>>(outp, lab, K0, V0, n);

  // 2) stable pair sort by output keys (4 x 8-bit passes), ends in K0/V0
  {
    const unsigned int *ks = K0, *vs = V0;
    unsigned int *kd = K1, *vd = V1;
    for (int p = 0; p < NPASS; p++) {
      int shift = 8 * p;
      k_hist<<<numTiles, THREADS, 0, stream>>>(ks, counts, n, shift, numTiles);
      run_scan(counts, bsums, L, stream);
      k_scatter<true><<<numTiles, STHREADS, 0, stream>>>(ks, vs, kd, vd, counts,
                                                         n, shift, numTiles);
      const unsigned int* t;
      t = ks; ks = kd; kd = (unsigned int*)t;
      t = vs; vs = vd; vd = (unsigned int*)t;
    }
  }

  // 3) dcg partials from gains-in-output-order (V0)
  const int perSweep = REDB * 8 * 64;
  const int iters = (n + perSweep - 1) / perSweep;
  k_dot<false><<<REDB, THREADS, 0, stream>>>(V0, pA, n, iters);

  // 4) sort gains descending (key-only); K0 is free now
  k_gkeys<<<mb, THREADS, 0, stream>>>(V0, K0, n);
  {
    const unsigned int* ks = K0;
    unsigned int* kd = K1;
    for (int p = 0; p < NPASS; p++) {
      int shift = 8 * p;
      k_hist<<<numTiles, THREADS, 0, stream>>>(ks, counts, n, shift, numTiles);
      run_scan(counts, bsums, L, stream);
      k_scatter<false><<<numTiles, STHREADS, 0, stream>>>(
          ks, (const unsigned int*)nullptr, kd, (unsigned int*)nullptr, counts,
          n, shift, numTiles);
      const unsigned int* t = ks; ks = kd; kd = (unsigned int*)t;
    }
  }

  // 5) zk partials from sorted gain keys (K0), then final ratio
  k_dot<true><<<REDB, THREADS, 0, stream>>>(K0, pB, n, iters);
  k_final<<<1, 32, 0, stream>>>(pA, pB, REDB, out);
}